// Electrostatics_32899449487756
// MI455X (gfx1250) — compile-verified
//
#include <hip/hip_runtime.h>

typedef __attribute__((ext_vector_type(2))) float v2f;
typedef __attribute__((ext_vector_type(8))) float v8f;

#define N_ATOMS        8192
#define NT             (N_ATOMS / 16)        // 512 tiles of 16 atoms
#define JT_PER_WAVE    8                     // j-tiles handled per wave
#define CHUNKS         (NT / JT_PER_WAVE)    // 64 chunks per i-tile row
#define WAVES_PER_BLK  8
#define THREADS        256
#define NBLOCKS        (NT * CHUNKS / WAVES_PER_BLK)  // 4096

#define KE_KCAL 332.0637f
#define LOG2E   1.442695040888963f

__global__ __launch_bounds__(THREADS)
void elec_tile_kernel(const float* __restrict__ q,
                      const float* __restrict__ xyz,
                      float* __restrict__ partial)
{
    const int lane = threadIdx.x & 31;
    const int wave = threadIdx.x >> 5;
    const int wgid = blockIdx.x * WAVES_PER_BLK + wave;   // global wave id
    const int i_tile = wgid >> 6;                         // wgid / CHUNKS
    const int chunk  = wgid & (CHUNKS - 1);
    const int m  = lane & 15;                             // row/col within tile
    const int hi = lane >> 4;                             // half-wave selector

    float acc = 0.0f;

    const int jt_begin = chunk * JT_PER_WAVE;
    const int jt_end   = jt_begin + JT_PER_WAVE;

    // Wave-uniform guard: EXEC stays all-ones inside (WMMA requirement).
    if (jt_end > i_tile) {
        // ---- i-tile: load positions, build A (scaled by -2) and C (= |x_i|^2 per row)
        const int ia = i_tile * 16 + m;
        const float xi = xyz[3 * ia + 0];
        const float yi = xyz[3 * ia + 1];
        const float zi = xyz[3 * ia + 2];
        const float ni_own = xi * xi + yi * yi + zi * zi;
        const float qi_own = q[ia];

        v2f a;
        a.x = hi ? (-2.0f * zi) : (-2.0f * xi);   // K = 2*hi + 0
        a.y = hi ? 0.0f         : (-2.0f * yi);   // K = 2*hi + 1

        v8f c;
        float qi_row[8];
        #pragma unroll
        for (int v = 0; v < 8; ++v) {
            const int src = v + 8 * hi;           // row M = v + 8*hi
            c[v]      = __shfl(ni_own, src, 32);
            qi_row[v] = __shfl(qi_own, src, 32);
        }

        int jt0 = jt_begin < i_tile ? i_tile : jt_begin;
        for (int jt = jt0; jt < jt_end; ++jt) {
            // ---- j-tile: build B, per-lane column norm / charge
            const int ja = jt * 16 + m;
            const float xj = xyz[3 * ja + 0];
            const float yj = xyz[3 * ja + 1];
            const float zj = xyz[3 * ja + 2];
            const float nj_own = xj * xj + yj * yj + zj * zj;
            const float kqj    = KE_KCAL * q[ja];

            v2f b;
            b.x = hi ? zj   : xj;                 // K = 2*hi + 0
            b.y = hi ? 0.0f : yj;                 // K = 2*hi + 1

            // D[M][N] = |x_i(M)|^2 - 2 * dot(x_i(M), x_j(N))
            v8f d = __builtin_amdgcn_wmma_f32_16x16x4_f32(
                false, a, false, b, (short)0, c, false, false);

            const int jg = jt * 16 + m;           // this lane's column atom
            #pragma unroll
            for (int v = 0; v < 8; ++v) {
                float d2 = d[v] + nj_own;
                d2 = fmaxf(d2, 1e-12f);
                const int ig = i_tile * 16 + v + 8 * hi;
                const bool valid = (jg > ig);

                // native, branchless: v_rsq / v_rcp / v_exp only
                const float inv_r  = __builtin_amdgcn_rsqf(d2);        // 1/r
                const float r      = d2 * inv_r;                       // sqrt(d2)
                const float inv_rh = __builtin_amdgcn_rsqf(d2 + 1.0f); // 1/sqrt(r^2+1)

                const float sarg = (r - 3.0f) * 0.5f;   // (r - r_on)/(r_off - r_on)
                const float t1   = 1.0f - sarg;
                // sigma(x) = x>0 ? exp(-1/x) : 0 ; exp(-1/x) = exp2(-log2e * rcp(x))
                float e1 = __builtin_amdgcn_exp2f(-LOG2E * __builtin_amdgcn_rcpf(t1));
                float e2 = __builtin_amdgcn_exp2f(-LOG2E * __builtin_amdgcn_rcpf(sarg));
                const float s1 = (t1   > 0.0f) ? e1 : 0.0f;
                const float s2 = (sarg > 0.0f) ? e2 : 0.0f;

                // fs*inv_rh + (1-fs)*inv_r == (s1*inv_rh + s2*inv_r) / (s1+s2)
                // s1+s2 >= 2*exp(-2) > 0 always, so raw rcp is safe.
                const float mix = (s1 * inv_rh + s2 * inv_r) *
                                  __builtin_amdgcn_rcpf(s1 + s2);
                const float e = kqj * qi_row[v] * mix;
                acc += valid ? e : 0.0f;
            }
        }
    }

    // ---- wave32 reduce, then block reduce via LDS; no atomics anywhere
    #pragma unroll
    for (int off = 16; off > 0; off >>= 1)
        acc += __shfl_down(acc, off, 32);

    __shared__ float sred[WAVES_PER_BLK];
    if (lane == 0) sred[wave] = acc;
    __syncthreads();
    if (threadIdx.x == 0) {
        float s = 0.0f;
        #pragma unroll
        for (int w = 0; w < WAVES_PER_BLK; ++w) s += sred[w];
        partial[blockIdx.x] = s;
    }
}

__global__ __launch_bounds__(THREADS)
void reduce_partials_kernel(const float* __restrict__ partial,
                            float* __restrict__ out, int n)
{
    float s = 0.0f;
    for (int i = threadIdx.x; i < n; i += THREADS) s += partial[i];
    #pragma unroll
    for (int off = 16; off > 0; off >>= 1)
        s += __shfl_down(s, off, 32);
    __shared__ float sred[THREADS / 32];
    const int wave = threadIdx.x >> 5;
    const int lane = threadIdx.x & 31;
    if (lane == 0) sred[wave] = s;
    __syncthreads();
    if (threadIdx.x == 0) {
        float t = 0.0f;
        #pragma unroll
        for (int w = 0; w < THREADS / 32; ++w) t += sred[w];
        out[0] = t;
    }
}

extern "C" void kernel_launch(void* const* d_in, const int* in_sizes, int n_in,
                              void* d_out, int out_size, void* d_ws, size_t ws_size,
                              hipStream_t stream) {
    const float* q   = (const float*)d_in[0];   // [8192]
    const float* xyz = (const float*)d_in[1];   // [8192, 3]
    float* partial = (float*)d_ws;              // NBLOCKS floats (16 KB)

    elec_tile_kernel<<<NBLOCKS, THREADS, 0, stream>>>(q, xyz, partial);
    reduce_partials_kernel<<<1, THREADS, 0, stream>>>(partial, (float*)d_out, NBLOCKS);
}